// LanguageAsDisturbance_25890062860792
// MI455X (gfx1250) — compile-verified
//
#include <hip/hip_runtime.h>
#include <hip/hip_bf16.h>

#define VOCABN 256
#define DSTATE 256
#define BATCH  32
#define TLEN   2048
#define SDIM   512
#define CHUNKS 64
#define CHUNK_LEN 32          // CHUNKS * CHUNK_LEN == TLEN
#define DT_F   0.1f
#define OUTBLK (BATCH * TLEN * DSTATE)   // 16777216 floats per output tensor

typedef __attribute__((ext_vector_type(2))) float v2f;
typedef __attribute__((ext_vector_type(4))) float v4f;
typedef __attribute__((ext_vector_type(8))) float v8f;

// D = A(16x4, f32) * B(4x16, f32) + C  -> v_wmma_f32_16x16x4_f32
__device__ __forceinline__ v8f wmma_f32_4(v2f a, v2f b, v8f c) {
  return __builtin_amdgcn_wmma_f32_16x16x4_f32(false, a, false, b, (short)0, c,
                                               false, false);
}

// ---------------------------------------------------------------------------
// Generic one-wave-per-16x16-tile fp32 WMMA GEMM:
//   C[M,N] = alpha * A(MxK) * op(B)   (+ diag_add on the diagonal)
//   TRANS_B=1: op(B)[k][n] = B[n][k]  (B stored N x K row-major)
// ---------------------------------------------------------------------------
template <bool TRANS_B>
__global__ __launch_bounds__(32) void gemm_tile_kernel(
    const float* __restrict__ A, const float* __restrict__ B,
    float* __restrict__ C, int Kdim, int lda, int ldb, int ldc, float alpha,
    float diag_add) {
  const int tn = blockIdx.x;
  const int tm = blockIdx.y;
  const int lane = threadIdx.x & 31;
  const int mrow = lane & 15;
  const int koff = (lane >> 4) << 1;   // 0 or 2
  const int ncol = lane & 15;
  const int rbase = (lane >> 4) << 3;  // 0 or 8
  const int arow = tm * 16 + mrow;
  const int bcol = tn * 16 + ncol;

  v8f acc = {};
  for (int k0 = 0; k0 < Kdim; k0 += 4) {
    v2f a, b;
    a = *(const v2f*)(A + arow * lda + k0 + koff);   // contiguous -> b64
    if (TRANS_B) {
      b = *(const v2f*)(B + bcol * ldb + k0 + koff); // contiguous -> b64
    } else {
      b.x = B[(k0 + koff) * ldb + bcol];
      b.y = B[(k0 + koff + 1) * ldb + bcol];
    }
    acc = wmma_f32_4(a, b, acc);
  }
#pragma unroll
  for (int r = 0; r < 8; ++r) {
    const int orow = tm * 16 + rbase + r;
    const int ocol = tn * 16 + ncol;
    float v = acc[r] * alpha;
    if (orow == ocol) v += diag_add;
    C[orow * ldc + ocol] = v;
  }
}

// ---------------------------------------------------------------------------
// Gauss-Jordan inversion of 256x256 SPD matrix (no pivoting needed).
// ---------------------------------------------------------------------------
__global__ __launch_bounds__(1024) void gauss_jordan_kernel(
    const float* __restrict__ Nmat, float* __restrict__ Ninv,
    float* __restrict__ aug) {
  const int n = DSTATE;
  const int w = 2 * n;
  const int tid = threadIdx.x;
  const int nt = blockDim.x;
  __shared__ float prow[2 * DSTATE];
  __shared__ float fac[DSTATE];
  __shared__ float pivinv;

  for (int idx = tid; idx < n * w; idx += nt) {
    const int r = idx / w, c = idx % w;
    aug[idx] = (c < n) ? Nmat[r * n + c] : ((c - n == r) ? 1.f : 0.f);
  }
  __syncthreads();

  for (int p = 0; p < n; ++p) {
    if (tid == 0) pivinv = 1.f / aug[p * w + p];
    __syncthreads();
    for (int c = tid; c < w; c += nt) {
      const float v = aug[p * w + c] * pivinv;
      prow[c] = v;
      aug[p * w + c] = v;
    }
    for (int r = tid; r < n; r += nt) fac[r] = aug[r * w + p];
    __syncthreads();
    for (int idx = tid; idx < n * w; idx += nt) {
      const int r = idx / w, c = idx % w;
      if (r != p) aug[idx] -= fac[r] * prow[c];
    }
    __syncthreads();
  }
  for (int idx = tid; idx < n * n; idx += nt) {
    const int r = idx / n, c = idx % n;
    Ninv[idx] = aug[r * w + n + c];
  }
}

// ---------------------------------------------------------------------------
// Build the 512x512 one-step transition matrix M (row-vector convention).
// ---------------------------------------------------------------------------
__global__ void build_M_kernel(const float* __restrict__ KNi,
                               const float* __restrict__ DNi,
                               float* __restrict__ M) {
  const int j = blockIdx.x * blockDim.x + threadIdx.x;  // 0..511
  const int i = blockIdx.y;                             // 0..511
  const int ii = i & (DSTATE - 1), jj = j & (DSTATE - 1);
  const float del = (ii == jj) ? 1.f : 0.f;
  const float kv = KNi[ii * DSTATE + jj];
  const float dv = DNi[ii * DSTATE + jj];
  float val;
  if (i < DSTATE && j < DSTATE)      val = del - DT_F * DT_F * kv;     // x -> x'
  else if (i < DSTATE)               val = -DT_F * kv;                  // x -> v'
  else if (j < DSTATE)               val = DT_F * (del - DT_F * dv);    // v -> x'
  else                               val = del - DT_F * dv;             // v -> v'
  M[i * SDIM + j] = val;
}

// ---------------------------------------------------------------------------
// Pack a 512x512 row-major matrix into WMMA B-fragment order:
//   Mp[(kg*SDIM + j)*4 + kk] = M[(kg*4 + kk)*SDIM + j]
// so a lane's (k, k+1) fragment at column j is one contiguous b64 load and a
// wave's fragment loads cover contiguous, fully-coalesced 16B blocks.
// ---------------------------------------------------------------------------
__global__ void pack_B_kernel(const float* __restrict__ M,
                              float* __restrict__ Mp) {
  const int j = blockIdx.x * blockDim.x + threadIdx.x;  // 0..511
  const int kg = blockIdx.y;                            // 0..127
  v4f p;
  p.x = M[(kg * 4 + 0) * SDIM + j];
  p.y = M[(kg * 4 + 1) * SDIM + j];
  p.z = M[(kg * 4 + 2) * SDIM + j];
  p.w = M[(kg * 4 + 3) * SDIM + j];
  *(v4f*)(Mp + (long)(kg * SDIM + j) * 4) = p;   // 16B store
}

// EG[tok][j] = dist_scale * (j<256 ? DT^2 : DT) * (emb @ Ninv)[tok][j&255]
__global__ void build_EG_kernel(const float* __restrict__ ENi,
                                const float* __restrict__ dist_scale,
                                float* __restrict__ EG) {
  const int j = blockIdx.x * blockDim.x + threadIdx.x;  // 0..511
  const int t = blockIdx.y;                             // 0..255
  const float sc = dist_scale[0] * ((j < DSTATE) ? (DT_F * DT_F) : DT_F);
  EG[t * SDIM + j] = sc * ENi[t * DSTATE + (j & (DSTATE - 1))];
}

// F_seq output: emb[tokens] * dist_scale, 16-byte vectorized
__global__ void gather_F_kernel(const int* __restrict__ tokens,
                                const float* __restrict__ emb,
                                const float* __restrict__ dist_scale,
                                float* __restrict__ Fout) {
  const long idx = (long)blockIdx.x * blockDim.x + threadIdx.x;  // OUTBLK/4
  const long bt = idx >> 6;
  const int dd4 = (int)(idx & 63);
  const float sc = dist_scale[0];
  const v4f e = *(const v4f*)(emb + (long)tokens[bt] * DSTATE + dd4 * 4);
  v4f o;
  o.x = e.x * sc; o.y = e.y * sc; o.z = e.z * sc; o.w = e.w * sc;
  *(v4f*)(Fout + idx * 4) = o;
}

// ---------------------------------------------------------------------------
// Chunked scan: one workgroup (16 waves) per chunk; state s (32x512) in LDS.
// Per step: s = s @ M + EG[tokens[:,t]].
// Wave w owns the 2x2 tile block {tm=0,1} x {tn=2w,2w+1}; B fragments come
// from the packed Mp stream (one b64 per fragment, coalesced per wave).
// pass 0: start from zero, emit chunk-end state E[c].
// pass 1: start from S0[c]; stream x_seq/v_seq (contiguous, v = x + OUTBLK).
// ---------------------------------------------------------------------------
__global__ __launch_bounds__(512) void scan_chunk_kernel(
    const float* __restrict__ Mp, const float* __restrict__ EG,
    const int* __restrict__ tokens, const float* __restrict__ S0,
    float* __restrict__ Eout, float* __restrict__ xv_out, int pass) {
  __shared__ float s[BATCH * SDIM];  // 64 KB
  const int c = blockIdx.x;
  const int tid = threadIdx.x;
  const int wave = tid >> 5;
  const int lane = tid & 31;
  const int tna = wave << 1;
  const int tnb = tna + 1;
  const int mrow = lane & 15;
  const int koff = (lane >> 4) << 1;
  const int ncol = lane & 15;
  const int rbase = (lane >> 4) << 3;
  const int jna = tna * 16 + ncol;
  const int jnb = tnb * 16 + ncol;

  if (pass == 0) {
    for (int i = tid; i < BATCH * SDIM; i += blockDim.x) s[i] = 0.f;
  } else {
    const float* s0 = S0 + (long)c * BATCH * SDIM;
    for (int i = tid; i < BATCH * SDIM; i += blockDim.x) s[i] = s0[i];
  }
  __syncthreads();

  const int t0 = c * CHUNK_LEN;
  for (int t = t0; t < t0 + CHUNK_LEN; ++t) {
    int tok0[8], tok1[8];
#pragma unroll
    for (int r = 0; r < 8; ++r) {
      tok0[r] = tokens[(rbase + r) * TLEN + t];
      tok1[r] = tokens[(16 + rbase + r) * TLEN + t];
    }

    v8f a00, a01, a10, a11;  // acc[tm][tn-sel]
#pragma unroll
    for (int r = 0; r < 8; ++r) {
      a00[r] = EG[tok0[r] * SDIM + jna];
      a01[r] = EG[tok0[r] * SDIM + jnb];
      a10[r] = EG[tok1[r] * SDIM + jna];
      a11[r] = EG[tok1[r] * SDIM + jnb];
    }

    for (int k0 = 0; k0 < SDIM; k0 += 4) {
      const float* bp = Mp + (long)(k0 >> 2) * SDIM * 4;
      __builtin_prefetch(bp + 4 * SDIM * 4 + jna * 4, 0, 3);  // 4 k-groups ahead
      v2f af0, af1, bf0, bf1;
      af0 = *(const v2f*)(s + mrow * SDIM + k0 + koff);
      af1 = *(const v2f*)(s + (16 + mrow) * SDIM + k0 + koff);
      bf0 = *(const v2f*)(bp + jna * 4 + koff);   // one b64, coalesced
      bf1 = *(const v2f*)(bp + jnb * 4 + koff);
      a00 = wmma_f32_4(af0, bf0, a00);
      a10 = wmma_f32_4(af1, bf0, a10);
      a01 = wmma_f32_4(af0, bf1, a01);
      a11 = wmma_f32_4(af1, bf1, a11);
    }
    __syncthreads();
#pragma unroll
    for (int r = 0; r < 8; ++r) {
      s[(rbase + r) * SDIM + jna]      = a00[r];
      s[(rbase + r) * SDIM + jnb]      = a01[r];
      s[(16 + rbase + r) * SDIM + jna] = a10[r];
      s[(16 + rbase + r) * SDIM + jnb] = a11[r];
    }
    if (pass == 1) {
      // branch-free x/v split: v_seq sits OUTBLK floats after x_seq
      const long shifta = (long)(jna >> 8) * (OUTBLK - DSTATE);
      const long shiftb = (long)(jnb >> 8) * (OUTBLK - DSTATE);
#pragma unroll
      for (int r = 0; r < 8; ++r) {
        const long row0 = (long)(rbase + r) * TLEN + t;
        const long row1 = (long)(16 + rbase + r) * TLEN + t;
        xv_out[shifta + row0 * DSTATE + jna] = a00[r];
        xv_out[shiftb + row0 * DSTATE + jnb] = a01[r];
        xv_out[shifta + row1 * DSTATE + jna] = a10[r];
        xv_out[shiftb + row1 * DSTATE + jnb] = a11[r];
      }
    }
    __syncthreads();
  }

  if (pass == 0) {
    float* e = Eout + (long)c * BATCH * SDIM;
    for (int i = tid; i < BATCH * SDIM; i += blockDim.x) e[i] = s[i];
  }
}

// ---------------------------------------------------------------------------
// Sequential prefix over chunks:  S0[0] = [x0,v0];  S0[c] = S0[c-1]@M^L + E[c-1]
// MLp is the packed M^CHUNK_LEN.
// ---------------------------------------------------------------------------
__global__ __launch_bounds__(512) void prefix_kernel(
    const float* __restrict__ MLp, const float* __restrict__ E,
    float* __restrict__ S0, const float* __restrict__ x0,
    const float* __restrict__ v0) {
  __shared__ float s[BATCH * SDIM];
  const int tid = threadIdx.x;
  const int wave = tid >> 5;
  const int lane = tid & 31;
  const int tna = wave << 1;
  const int tnb = tna + 1;
  const int mrow = lane & 15;
  const int koff = (lane >> 4) << 1;
  const int ncol = lane & 15;
  const int rbase = (lane >> 4) << 3;
  const int jna = tna * 16 + ncol;
  const int jnb = tnb * 16 + ncol;

  for (int i = tid; i < BATCH * SDIM; i += blockDim.x) {
    const int j = i & (SDIM - 1);
    const float v = (j < DSTATE) ? x0[j] : v0[j - DSTATE];
    s[i] = v;
    S0[i] = v;  // chunk 0 start state
  }
  __syncthreads();

  for (int c = 1; c < CHUNKS; ++c) {
    const float* e = E + (long)(c - 1) * BATCH * SDIM;
    v8f a00, a01, a10, a11;
#pragma unroll
    for (int r = 0; r < 8; ++r) {
      a00[r] = e[(rbase + r) * SDIM + jna];
      a01[r] = e[(rbase + r) * SDIM + jnb];
      a10[r] = e[(16 + rbase + r) * SDIM + jna];
      a11[r] = e[(16 + rbase + r) * SDIM + jnb];
    }
    for (int k0 = 0; k0 < SDIM; k0 += 4) {
      const float* bp = MLp + (long)(k0 >> 2) * SDIM * 4;
      v2f af0, af1, bf0, bf1;
      af0 = *(const v2f*)(s + mrow * SDIM + k0 + koff);
      af1 = *(const v2f*)(s + (16 + mrow) * SDIM + k0 + koff);
      bf0 = *(const v2f*)(bp + jna * 4 + koff);
      bf1 = *(const v2f*)(bp + jnb * 4 + koff);
      a00 = wmma_f32_4(af0, bf0, a00);
      a10 = wmma_f32_4(af1, bf0, a10);
      a01 = wmma_f32_4(af0, bf1, a01);
      a11 = wmma_f32_4(af1, bf1, a11);
    }
    __syncthreads();
    float* s0c = S0 + (long)c * BATCH * SDIM;
#pragma unroll
    for (int r = 0; r < 8; ++r) {
      const int o00 = (rbase + r) * SDIM + jna;
      const int o01 = (rbase + r) * SDIM + jnb;
      const int o10 = (16 + rbase + r) * SDIM + jna;
      const int o11 = (16 + rbase + r) * SDIM + jnb;
      s[o00] = a00[r]; s0c[o00] = a00[r];
      s[o01] = a01[r]; s0c[o01] = a01[r];
      s[o10] = a10[r]; s0c[o10] = a10[r];
      s[o11] = a11[r]; s0c[o11] = a11[r];
    }
    __syncthreads();
  }
}

// ---------------------------------------------------------------------------
// logits = x_seq @ dec_w^T + dec_b    (65536 x 256) @ (256 x 256)^T
// ---------------------------------------------------------------------------
__global__ __launch_bounds__(32) void logits_kernel(
    const float* __restrict__ X, const float* __restrict__ W,
    const float* __restrict__ bias, float* __restrict__ out) {
  const int tn = blockIdx.x;   // vocab tile 0..15
  const int tm = blockIdx.y;   // row tile 0..4095
  const int lane = threadIdx.x & 31;
  const int mrow = lane & 15;
  const int koff = (lane >> 4) << 1;
  const int ncol = lane & 15;
  const int rbase = (lane >> 4) << 3;
  const int col = tn * 16 + ncol;

  v8f acc;
  const float bv = bias[col];
#pragma unroll
  for (int r = 0; r < 8; ++r) acc[r] = bv;

  const long arow = (long)tm * 16 + mrow;
  for (int k0 = 0; k0 < DSTATE; k0 += 4) {
    v2f a, b;
    a = *(const v2f*)(X + arow * DSTATE + k0 + koff);  // b64
    b = *(const v2f*)(W + (long)col * DSTATE + k0 + koff);  // b64
    acc = wmma_f32_4(a, b, acc);
  }
#pragma unroll
  for (int r = 0; r < 8; ++r) {
    const long orow = (long)tm * 16 + rbase + r;
    out[orow * VOCABN + tn * 16 + ncol] = acc[r];
  }
}

// ---------------------------------------------------------------------------
// Launcher
// ---------------------------------------------------------------------------
extern "C" void kernel_launch(void* const* d_in, const int* in_sizes, int n_in,
                              void* d_out, int out_size, void* d_ws,
                              size_t ws_size, hipStream_t stream) {
  const int* tokens = (const int*)d_in[0];
  const float* N_raw = (const float*)d_in[1];
  const float* D_raw = (const float*)d_in[2];
  const float* K_raw = (const float*)d_in[3];
  const float* emb = (const float*)d_in[4];
  const float* dist_scale = (const float*)d_in[5];
  const float* dec_w = (const float*)d_in[6];
  const float* dec_b = (const float*)d_in[7];
  const float* x0 = (const float*)d_in[8];
  const float* v0 = (const float*)d_in[9];

  float* out_logits = (float*)d_out;
  float* out_x = out_logits + (long)OUTBLK;
  float* out_F = out_logits + 3L * OUTBLK;

  float* ws = (float*)d_ws;
  const int d2 = DSTATE * DSTATE;  // 65536
  const int s2 = SDIM * SDIM;      // 262144
  float* Nmat = ws;
  float* Dmat = Nmat + d2;
  float* Kmat = Dmat + d2;
  float* Ninv = Kmat + d2;
  float* KNi = Ninv + d2;
  float* DNi = KNi + d2;
  float* ENi = DNi + d2;
  float* Mmat = ENi + d2;          // 512*512
  float* P0 = Mmat + s2;           // 512*512
  float* P1 = P0 + s2;             // 512*512
  float* Mp = P1 + s2;             // 512*512 packed step matrix
  float* MLp = Mp + s2;            // 512*512 packed chunk matrix
  float* EG = MLp + s2;            // 256*512
  float* aug = EG + VOCABN * SDIM; // 256*512
  float* Ech = aug + DSTATE * 2 * DSTATE;      // CHUNKS*32*512
  float* S0 = Ech + (long)CHUNKS * BATCH * SDIM;

  const dim3 g16(DSTATE / 16, DSTATE / 16);
  const dim3 g32(SDIM / 16, SDIM / 16);

  // 1) SPD matrices: R @ R^T (+ diag)
  gemm_tile_kernel<true><<<g16, 32, 0, stream>>>(N_raw, N_raw, Nmat, DSTATE,
                                                 DSTATE, DSTATE, DSTATE, 1.f, 0.1f);
  gemm_tile_kernel<true><<<g16, 32, 0, stream>>>(D_raw, D_raw, Dmat, DSTATE,
                                                 DSTATE, DSTATE, DSTATE, 1.f, 0.f);
  gemm_tile_kernel<true><<<g16, 32, 0, stream>>>(K_raw, K_raw, Kmat, DSTATE,
                                                 DSTATE, DSTATE, DSTATE, 1.f, 0.01f);
  // 2) N^-1
  gauss_jordan_kernel<<<1, 1024, 0, stream>>>(Nmat, Ninv, aug);
  // 3) K@Ninv, D@Ninv, emb@Ninv
  gemm_tile_kernel<false><<<g16, 32, 0, stream>>>(Kmat, Ninv, KNi, DSTATE,
                                                  DSTATE, DSTATE, DSTATE, 1.f, 0.f);
  gemm_tile_kernel<false><<<g16, 32, 0, stream>>>(Dmat, Ninv, DNi, DSTATE,
                                                  DSTATE, DSTATE, DSTATE, 1.f, 0.f);
  gemm_tile_kernel<false><<<g16, 32, 0, stream>>>(emb, Ninv, ENi, DSTATE,
                                                  DSTATE, DSTATE, DSTATE, 1.f, 0.f);
  // 4) transition matrix M, input-injection table EG
  build_M_kernel<<<dim3(2, SDIM), 256, 0, stream>>>(KNi, DNi, Mmat);
  build_EG_kernel<<<dim3(2, VOCABN), 256, 0, stream>>>(ENi, dist_scale, EG);
  // 5) M^CHUNK_LEN (=M^32) by repeated squaring
  gemm_tile_kernel<false><<<g32, 32, 0, stream>>>(Mmat, Mmat, P0, SDIM, SDIM,
                                                  SDIM, SDIM, 1.f, 0.f);  // M^2
  gemm_tile_kernel<false><<<g32, 32, 0, stream>>>(P0, P0, P1, SDIM, SDIM,
                                                  SDIM, SDIM, 1.f, 0.f);  // M^4
  gemm_tile_kernel<false><<<g32, 32, 0, stream>>>(P1, P1, P0, SDIM, SDIM,
                                                  SDIM, SDIM, 1.f, 0.f);  // M^8
  gemm_tile_kernel<false><<<g32, 32, 0, stream>>>(P0, P0, P1, SDIM, SDIM,
                                                  SDIM, SDIM, 1.f, 0.f);  // M^16
  gemm_tile_kernel<false><<<g32, 32, 0, stream>>>(P1, P1, P0, SDIM, SDIM,
                                                  SDIM, SDIM, 1.f, 0.f);  // M^32
  // 5b) pack step matrix and chunk matrix into WMMA B-fragment order
  pack_B_kernel<<<dim3(2, SDIM / 4), 256, 0, stream>>>(Mmat, Mp);
  pack_B_kernel<<<dim3(2, SDIM / 4), 256, 0, stream>>>(P0, MLp);
  // 6) F_seq output (pure gather, b128 path)
  gather_F_kernel<<<OUTBLK / 4 / 256, 256, 0, stream>>>(tokens, emb,
                                                        dist_scale, out_F);
  // 7) pass 1: chunk-local scans from zero state
  scan_chunk_kernel<<<CHUNKS, 512, 0, stream>>>(Mp, EG, tokens, nullptr, Ech,
                                                nullptr, 0);
  // 8) sequential prefix over chunk boundary states
  prefix_kernel<<<1, 512, 0, stream>>>(MLp, Ech, S0, x0, v0);
  // 9) pass 2: re-run chunks from true start states, emit x_seq / v_seq
  scan_chunk_kernel<<<CHUNKS, 512, 0, stream>>>(Mp, EG, tokens, S0, nullptr,
                                                out_x, 1);
  // 10) decode
  logits_kernel<<<dim3(VOCABN / 16, (BATCH * TLEN) / 16), 32, 0, stream>>>(
      out_x, dec_w, dec_b, out_logits);
}